// GCNSampling_18141941859035
// MI455X (gfx1250) — compile-verified
//
#include <hip/hip_runtime.h>

#define N0 100000
#define N1 25000
#define N2 5000
#define E0 400000
#define E1 80000
#define IN_FEATS 512
#define NH 512
#define NC 64

typedef __attribute__((ext_vector_type(2))) float v2f;
typedef __attribute__((ext_vector_type(4))) float v4f;
typedef __attribute__((ext_vector_type(8))) float v8f;

// ---------------------------------------------------------------------------
// WMMA f32 16x16x4 mainloop: one wave computes a 16 (M) x 64 (N) strip of
// C = A @ W^T, with A fragments reused across the 4 N-tiles per k-step.
//
// A-matrix 16x4 f32 fragment (ISA 7.12.2): lanes 0-15 hold row M=lane,
// K=(k,k+1) in VGPR0/1; lanes 16-31 hold row M=lane-16, K=(k+2,k+3).
// B-matrix 4x16 fragment mirrors this with B[k][n] = W[n][k], so both are a
// single contiguous v2f load from row-major memory.
// ---------------------------------------------------------------------------
__device__ __forceinline__ void wmma_mainloop(const float* __restrict__ A, int lda, int M,
                                              const float* __restrict__ W, int ldw,
                                              int mBase, int nBase, int K, v8f acc[4]) {
    const int lane = threadIdx.x & 31;
    const int half = lane >> 4;    // 0: K pair (k,k+1), 1: K pair (k+2,k+3)
    const int lr   = lane & 15;

    int m = mBase + lr;
    if (m > M - 1) m = M - 1;      // clamp loads; stores are guarded later

    const float* aRow = A + (size_t)m * lda + 2 * half;
    const float* w0 = W + (size_t)(nBase +  0 + lr) * ldw + 2 * half;
    const float* w1 = W + (size_t)(nBase + 16 + lr) * ldw + 2 * half;
    const float* w2 = W + (size_t)(nBase + 32 + lr) * ldw + 2 * half;
    const float* w3 = W + (size_t)(nBase + 48 + lr) * ldw + 2 * half;

#pragma unroll 4
    for (int k = 0; k < K; k += 4) {
        v2f a  = *(const v2f*)(aRow + k);
        v2f b0 = *(const v2f*)(w0 + k);
        v2f b1 = *(const v2f*)(w1 + k);
        v2f b2 = *(const v2f*)(w2 + k);
        v2f b3 = *(const v2f*)(w3 + k);
        acc[0] = __builtin_amdgcn_wmma_f32_16x16x4_f32(false, a, false, b0, (short)0, acc[0], false, false);
        acc[1] = __builtin_amdgcn_wmma_f32_16x16x4_f32(false, a, false, b1, (short)0, acc[1], false, false);
        acc[2] = __builtin_amdgcn_wmma_f32_16x16x4_f32(false, a, false, b2, (short)0, acc[2], false, false);
        acc[3] = __builtin_amdgcn_wmma_f32_16x16x4_f32(false, a, false, b3, (short)0, acc[3], false, false);
    }
}

// Stage 1: h = relu(feats @ W0^T + b0) - h0_hist        [N0, 512]
// Block = 8 waves; block owns 16 rows, wave w owns cols [w*64, w*64+64).
__global__ __launch_bounds__(256) void gemm0_kernel(const float* __restrict__ feats,
                                                    const float* __restrict__ W0,
                                                    const float* __restrict__ b0,
                                                    const float* __restrict__ h0_hist,
                                                    float* __restrict__ h) {
    const int wave = threadIdx.x >> 5;
    const int lane = threadIdx.x & 31;
    const int half = lane >> 4, lr = lane & 15;
    const int mBase = blockIdx.x * 16;
    const int nBase = wave * 64;

    v8f acc[4] = {};
    wmma_mainloop(feats, IN_FEATS, N0, W0, IN_FEATS, mBase, nBase, IN_FEATS, acc);

#pragma unroll
    for (int t = 0; t < 4; ++t) {
        const int n = nBase + t * 16 + lr;
        const float bias = b0[n];
#pragma unroll
        for (int v = 0; v < 8; ++v) {
            const int row = mBase + v + 8 * half;  // C/D layout: lanes>=16 hold M=v+8
            if (row < N0) {
                float val = acc[t][v] + bias;
                val = val > 0.f ? val : 0.f;
                h[(size_t)row * NH + n] = val - h0_hist[(size_t)row * NH + n];
            }
        }
    }
}

// Stage 2 GEMM: lin = A0 @ W1^T + b1; h1c = concat(lin, relu(lin)) - h1_hist  [N1, 1024]
__global__ __launch_bounds__(256) void gemm1_kernel(const float* __restrict__ A0,
                                                    const float* __restrict__ W1,
                                                    const float* __restrict__ b1,
                                                    const float* __restrict__ h1_hist,
                                                    float* __restrict__ h1c) {
    const int wave = threadIdx.x >> 5;
    const int lane = threadIdx.x & 31;
    const int half = lane >> 4, lr = lane & 15;
    const int mBase = blockIdx.x * 16;
    const int nBase = wave * 64;

    v8f acc[4] = {};
    wmma_mainloop(A0, NH, N1, W1, NH, mBase, nBase, NH, acc);

#pragma unroll
    for (int t = 0; t < 4; ++t) {
        const int n = nBase + t * 16 + lr;
        const float bias = b1[n];
#pragma unroll
        for (int v = 0; v < 8; ++v) {
            const int row = mBase + v + 8 * half;
            if (row < N1) {
                const size_t base = (size_t)row * (2 * NH);
                const float lin = acc[t][v] + bias;
                const float rel = lin > 0.f ? lin : 0.f;
                h1c[base + n]      = lin - h1_hist[base + n];
                h1c[base + NH + n] = rel - h1_hist[base + NH + n];
            }
        }
    }
}

// Stage 3 GEMM: out = A1 @ W2^T + b2     [N2, 64]; wave w owns rows blk*128+w*16.
__global__ __launch_bounds__(256) void gemm2_kernel(const float* __restrict__ A1,
                                                    const float* __restrict__ W2,
                                                    const float* __restrict__ b2,
                                                    float* __restrict__ out) {
    const int wave = threadIdx.x >> 5;
    const int lane = threadIdx.x & 31;
    const int half = lane >> 4, lr = lane & 15;
    const int mBase = blockIdx.x * 128 + wave * 16;
    if (mBase >= N2) return;  // wave-uniform exit keeps EXEC all-1s for live waves

    v8f acc[4] = {};
    wmma_mainloop(A1, 2 * NH, N2, W2, 2 * NH, mBase, 0, 2 * NH, acc);

#pragma unroll
    for (int t = 0; t < 4; ++t) {
        const int n = t * 16 + lr;
        const float bias = b2[n];
#pragma unroll
        for (int v = 0; v < 8; ++v) {
            const int row = mBase + v + 8 * half;
            if (row < N2) out[(size_t)row * NC + n] = acc[t][v] + bias;
        }
    }
}

// Edge scatter, D=512: one block per edge, 2 floats/thread; f32 atomics hit L2.
__global__ __launch_bounds__(256) void scatter0_kernel(const float* __restrict__ h,
                                                       const int* __restrict__ src,
                                                       const int* __restrict__ dst,
                                                       float* __restrict__ sums,
                                                       float* __restrict__ deg) {
    const int e = blockIdx.x;
    const int s = src[e], d = dst[e];
    const float* hp = h + (size_t)s * NH;
    float* sp = sums + (size_t)d * NH;
    const int i = threadIdx.x * 2;
    v2f v = *(const v2f*)(hp + i);
    atomicAdd(sp + i,     v.x);
    atomicAdd(sp + i + 1, v.y);
    if (threadIdx.x == 0) atomicAdd(deg + d, 1.0f);
}

// Edge scatter, D=1024: one block per edge, 4 floats/thread.
__global__ __launch_bounds__(256) void scatter1_kernel(const float* __restrict__ h,
                                                       const int* __restrict__ src,
                                                       const int* __restrict__ dst,
                                                       float* __restrict__ sums,
                                                       float* __restrict__ deg) {
    const int e = blockIdx.x;
    const int s = src[e], d = dst[e];
    const float* hp = h + (size_t)s * (2 * NH);
    float* sp = sums + (size_t)d * (2 * NH);
    const int i = threadIdx.x * 4;
    v4f v = *(const v4f*)(hp + i);
    atomicAdd(sp + i,     v.x);
    atomicAdd(sp + i + 1, v.y);
    atomicAdd(sp + i + 2, v.z);
    atomicAdd(sp + i + 3, v.w);
    if (threadIdx.x == 0) atomicAdd(deg + d, 1.0f);
}

// In-place: sums[i] = sums[i] / max(deg,1) + aggh[i]   (dShift = log2(D))
__global__ __launch_bounds__(256) void agg_kernel(float* __restrict__ sums,
                                                  const float* __restrict__ deg,
                                                  const float* __restrict__ aggh,
                                                  int total, int dShift) {
    const int i = blockIdx.x * blockDim.x + threadIdx.x;
    if (i >= total) return;
    const float dg = deg[i >> dShift];
    const float r = 1.0f / (dg > 1.0f ? dg : 1.0f);
    sums[i] = sums[i] * r + aggh[i];
}

extern "C" void kernel_launch(void* const* d_in, const int* in_sizes, int n_in,
                              void* d_out, int out_size, void* d_ws, size_t ws_size,
                              hipStream_t stream) {
    const float* feats   = (const float*)d_in[0];
    const float* h0_hist = (const float*)d_in[1];
    const float* agg_h0  = (const float*)d_in[2];
    const float* h1_hist = (const float*)d_in[3];
    const float* agg_h1  = (const float*)d_in[4];
    const float* W0      = (const float*)d_in[5];
    const float* b0      = (const float*)d_in[6];
    const float* W1      = (const float*)d_in[7];
    const float* b1      = (const float*)d_in[8];
    const float* W2      = (const float*)d_in[9];
    const float* b2      = (const float*)d_in[10];
    const int*   src0    = (const int*)d_in[11];
    const int*   dst0    = (const int*)d_in[12];
    const int*   src1    = (const int*)d_in[13];
    const int*   dst1    = (const int*)d_in[14];
    float* out = (float*)d_out;

    // Workspace layout (lifetimes overlaid):
    //   h region:  h [N0,512] f32 (stage 1), reused as h1c [N1,1024] (stage 2)
    //   sums0:     [N1,512], converted in place to A0
    //   sums1:     [N2,1024], converted in place to A1
    char* ws = (char*)d_ws;
    size_t off = 0;
    auto take = [&](size_t bytes) -> char* {
        char* p = ws + off;
        off = (off + bytes + 255) & ~(size_t)255;
        return p;
    };
    float* h     = (float*)take((size_t)N0 * NH * sizeof(float));      // >= N1*1024*4
    float* sums0 = (float*)take((size_t)N1 * NH * sizeof(float));
    float* deg0  = (float*)take((size_t)N1 * sizeof(float));
    float* sums1 = (float*)take((size_t)N2 * 2 * NH * sizeof(float));
    float* deg1  = (float*)take((size_t)N2 * sizeof(float));
    (void)ws_size; (void)in_sizes; (void)n_in; (void)out_size;

    // Zero accumulators every call (graph-capturable memset nodes).
    hipMemsetAsync(sums0, 0, (size_t)N1 * NH * sizeof(float), stream);
    hipMemsetAsync(deg0,  0, (size_t)N1 * sizeof(float), stream);
    hipMemsetAsync(sums1, 0, (size_t)N2 * 2 * NH * sizeof(float), stream);
    hipMemsetAsync(deg1,  0, (size_t)N2 * sizeof(float), stream);

    // Stage 1: fused GEMM + bias + relu + control-variate subtract.
    gemm0_kernel<<<N0 / 16, 256, 0, stream>>>(feats, W0, b0, h0_hist, h);

    // Stage 1 aggregation: gather/scatter-mean by edges.
    scatter0_kernel<<<E0, 256, 0, stream>>>(h, src0, dst0, sums0, deg0);
    agg_kernel<<<(N1 * NH + 255) / 256, 256, 0, stream>>>(sums0, deg0, agg_h0, N1 * NH, 9);

    // Stage 2: fused GEMM + bias + concat(relu) + subtract (h reused as h1c).
    gemm1_kernel<<<(N1 + 15) / 16, 256, 0, stream>>>(sums0, W1, b1, h1_hist, h);

    // Stage 2 aggregation.
    scatter1_kernel<<<E1, 256, 0, stream>>>(h, src1, dst1, sums1, deg1);
    agg_kernel<<<(N2 * 2 * NH + 255) / 256, 256, 0, stream>>>(sums1, deg1, agg_h1, N2 * 2 * NH, 10);

    // Stage 3: final GEMM + bias.
    gemm2_kernel<<<(N2 + 127) / 128, 256, 0, stream>>>(sums1, W2, b2, out);
}